// kernel_generated_2_21887153341253
// MI455X (gfx1250) — compile-verified
//
#include <hip/hip_runtime.h>

typedef __attribute__((ext_vector_type(2))) float v2f;
typedef __attribute__((ext_vector_type(8))) float v8f;

#define C_IN_G   24                 // input channels per group
#define M_G      96                 // output channels per group
#define W_IMG    56
#define H_IMG    56
#define KW       7
#define KPAD     8                  // taps padded 7 -> 8 so K' = c*8+k
#define KTOTP    (C_IN_G * KPAD)    // 192 = 48 WMMA steps of K=4
#define NSTEP    (KTOTP / 4)        // 48
#define XSTRIDE  72                 // padded LDS row stride (p+k reaches 70)
#define HW       (H_IMG * W_IMG)

// ---------------------------------------------------------------------------
// Prep kernel: pack weights into a pair-interleaved WMMA layout in d_ws:
//   apack[(K'>>1)*96 + m][j] = w[c, m, k]  (K' = c*8+k = 2*(K'>>1)+j, k==7 -> 0)
// A lane's two A-fragment values per WMMA step (K' = 4s+kb, 4s+kb+1) are
// adjacent -> one aligned global_load_b64 per step.
// 192*96 floats = 73728 bytes of workspace.
// ---------------------------------------------------------------------------
__global__ __launch_bounds__(256)
void pack_weights(const float* __restrict__ w, float* __restrict__ apack)
{
    int idx = blockIdx.x * 256 + threadIdx.x;      // [0, 192*96)
    if (idx >= KTOTP * M_G) return;
    const int Kp = idx / M_G;
    const int m  = idx - Kp * M_G;
    const int c  = Kp >> 3;
    const int k  = Kp & 7;
    float v = 0.0f;
    if (k < KW)
        v = w[c * (M_G * KW) + m * KW + k];
    apack[((Kp >> 1) * M_G + m) * 2 + (Kp & 1)] = v;
}

// ---------------------------------------------------------------------------
// Main kernel: one block per (row h, group g). GEMM per row:
//   Y[m, p] = sum_{K'} A[K'][m] * Xrow[c, p + k - 3]
// M = 96 (6 tiles), N = 56 (4 tiles, padded), K' = 192 (48 x v_wmma k=4).
// ---------------------------------------------------------------------------
__global__ __launch_bounds__(256)
void conv1x7_grouped_wmma(const float* __restrict__ x,
                          const float* __restrict__ apack,
                          const int*   __restrict__ shiftp,
                          float*       __restrict__ out)
{
    __shared__ float xlds[C_IN_G * XSTRIDE];

    const int h    = blockIdx.x;          // image row
    const int g    = blockIdx.y;          // group (0 or 1)
    const int tid  = threadIdx.x;
    const int lane = tid & 31;            // wave32
    const int wave = tid >> 5;            // 8 waves per block

    // Stage padded input row: xlds[c][j] = x[g*24+c, h, j-3] (zeros outside)
    for (int j = tid; j < C_IN_G * XSTRIDE; j += 256) {
        const int c   = j / XSTRIDE;
        const int col = j - c * XSTRIDE;
        const int ww  = col - 3;
        float v = 0.0f;
        if (ww >= 0 && ww < W_IMG)
            v = x[(size_t)(g * C_IN_G + c) * HW + (size_t)h * W_IMG + ww];
        xlds[j] = v;
    }
    __syncthreads();

    // roll(shift, axis=H): conv row h lands in output row (h+shift) mod 56
    const int shift = shiftp[0];
    int hout = (h + shift) % H_IMG;
    if (hout < 0) hout += H_IMG;

    const int ml = lane & 15;             // M/N position within tile
    const int kb = (lane >> 4) << 1;      // per-lane K-pair base: 0 or 2

    // 24 tiles = 6 mtiles x 4 ntiles; 8 waves -> 3 tiles each
    for (int t = wave; t < 24; t += 8) {
        const int mtile = t >> 2;
        const int ntile = t & 3;
        const int mbase = mtile * 16;
        const int pbase = ntile * 16;

        // Per-lane base pointers; all inner-loop offsets are compile-time.
        // A pair index for step s: 2s + kb/2  ->  v2f stride 192 per step.
        const v2f*   ap    = (const v2f*)apack + (kb >> 1) * M_G + mbase + ml;
        const float* xbase = xlds + pbase + ml + kb;

        v8f acc = {0.f, 0.f, 0.f, 0.f, 0.f, 0.f, 0.f, 0.f};

#pragma unroll
        for (int s = 0; s < NSTEP; ++s) {
            // c = s>>1 (kb/+1 never carries past a multiple of 8), k = (s&1)*4 + kb
            const int xoff = (s >> 1) * XSTRIDE + ((s & 1) << 2);
            v2f a = ap[s * (2 * M_G)];        // {A[4s+kb][m], A[4s+kb+1][m]}, b64
            v2f b;
            b.x = xbase[xoff];                // Xrow[c][p + k - 3]
            b.y = xbase[xoff + 1];            // k+1 (k==7 pairs with zero A col)
            acc = __builtin_amdgcn_wmma_f32_16x16x4_f32(
                false, a, false, b, (short)0, acc, false, false);
        }

        // D layout: VGPR r -> M = mbase + r (+8 for lanes 16-31), N = lane&15
        const int n = pbase + ml;
        if (n < W_IMG) {
            const int mhalf = (lane >> 4) << 3;   // 0 or 8
            float* op = out + (size_t)(g * M_G) * HW + (size_t)hout * W_IMG + n;
#pragma unroll
            for (int r = 0; r < 8; ++r) {
                const int m = mbase + r + mhalf;
                op[(size_t)m * HW] = acc[r];
            }
        }
    }
}

extern "C" void kernel_launch(void* const* d_in, const int* in_sizes, int n_in,
                              void* d_out, int out_size, void* d_ws, size_t ws_size,
                              hipStream_t stream) {
    const float* x      = (const float*)d_in[0];   // (1, 48, 56, 56) f32
    const float* w      = (const float*)d_in[1];   // (24, 96, 7) f32
    const int*   shiftp = (const int*)d_in[2];     // scalar shift
    float*       out    = (float*)d_out;           // (1, 192, 56, 56) f32
    float*       apack  = (float*)d_ws;            // 192*96 floats = 73728 B

    // Pack weights (zero-padded K'=c*8+k, pair-interleaved) into workspace.
    const int nPack = KTOTP * M_G;                 // 18432
    hipLaunchKernelGGL(pack_weights, dim3((nPack + 255) / 256), dim3(256),
                       0, stream, w, apack);

    // Main conv-as-GEMM kernel.
    dim3 grid(H_IMG, 2);   // 56 rows x 2 groups
    dim3 block(256);       // 8 wave32
    hipLaunchKernelGGL(conv1x7_grouped_wmma, grid, block, 0, stream,
                       x, apack, shiftp, out);
}